// EquivariantUpSampling_72773925864032
// MI455X (gfx1250) — compile-verified
//
#include <hip/hip_runtime.h>

// EquivariantUpSampling: periodic rot90 (k = p[b,2] in {0,1}) + one-hot scatter
// into channel-slot p[b,2], spatial 2x2 block position (p[b,0], p[b,1]).
// x: (8,128,32,32) f32  ->  out: (8,256,64,64) f32 (mostly zeros).
//
// Memory-bound (36 MiB total traffic, 0 FLOPs). CDNA5 path used:
//   - global_load_async_to_lds_b128 (ASYNCcnt) stages each 4 KiB input plane
//     into LDS so the k=1 transposed gather is an LDS column read (pitch 36
//     floats: 16B-aligned rows for B128 deposits, reduced bank conflicts),
//   - s_wait_asynccnt 0 before the workgroup barrier,
//   - all global writes are coalesced float4 stores in one output pass.

#define B_SZ     8
#define C_IN     128
#define HW       32
#define LDS_PITCH 36   // 32 + 4 pad: rows stay 16B-aligned, breaks conflicts on column reads

__global__ __launch_bounds__(256) void equiv_upsample_kernel(
    const float* __restrict__ x,
    const int*   __restrict__ p,
    float*       __restrict__ out)
{
    __shared__ __align__(16) float tile[HW * LDS_PITCH];

    const int ch = blockIdx.x;   // input channel 0..127
    const int b  = blockIdx.y;   // batch 0..7
    const int t  = threadIdx.x;  // 0..255

    const int p0 = p[b * 3 + 0]; // H slot (0/1)
    const int p1 = p[b * 3 + 1]; // W slot (0/1)
    const int p2 = p[b * 3 + 2]; // rotation k AND s_H channel slot (0/1)

    // ---- stage the (b,ch) 32x32 plane into LDS via async B128 deposits ----
    const float* src = x + (((size_t)(b * C_IN + ch)) << 10); // * 1024 floats
    {
        const int row  = t >> 3;   // 0..31
        const int col4 = t & 7;    // float4 index within row, 0..7
        const float* g = src + row * HW + col4 * 4;
        // low 32 bits of a flat shared-aperture address == LDS byte offset
        unsigned lds_off = (unsigned)(size_t)(void*)(tile + row * LDS_PITCH + col4 * 4);
        asm volatile("global_load_async_to_lds_b128 %0, %1, off"
                     :: "v"(lds_off), "v"(g)
                     : "memory");
    }
    asm volatile("s_wait_asynccnt 0x0" ::: "memory");
    __syncthreads();

    // ---- single pass over the two output planes for this input channel ----
    // out planes (b, 2*ch + s), s in {0,1}; plane s != p2 is all zero.
    float4* oplane = (float4*)(out + (((size_t)(b * 256 + ch * 2)) << 12)); // * 4096 floats

    #pragma unroll
    for (int j = 0; j < 8; ++j) {
        const int q     = j * 256 + t;   // quad id, 0..2047 (2 planes x 1024 float4)
        const int plane = q >> 10;       // 0 or 1 (uniform per iteration)
        const int qq    = q & 1023;
        const int Hrow  = qq >> 4;       // output row 0..63
        const int qx    = qq & 15;       // float4 within row (W = 4*qx .. 4*qx+3)

        float4 v4 = make_float4(0.f, 0.f, 0.f, 0.f);

        if (plane == p2 && (Hrow & 1) == p0) {
            const int u  = Hrow >> 1;    // source row 0..31
            const int v0 = 2 * qx;       // source cols for elements {0,1} / {2,3}
            const int v1 = v0 + 1;
            float a, c;
            if (p2 == 0) {               // identity
                a = tile[u * LDS_PITCH + v0];
                c = tile[u * LDS_PITCH + v1];
            } else {                     // 90-deg CCW with periodic wrap: x[v, (32-u)&31]
                const int uu = (HW - u) & (HW - 1);
                a = tile[v0 * LDS_PITCH + uu];
                c = tile[v1 * LDS_PITCH + uu];
            }
            // within the float4, only lanes with (W&1)==p1 are nonzero
            v4.x = (p1 == 0) ? a : 0.f;
            v4.y = (p1 == 1) ? a : 0.f;
            v4.z = (p1 == 0) ? c : 0.f;
            v4.w = (p1 == 1) ? c : 0.f;
        }
        oplane[q] = v4;  // fully coalesced 16B store
    }
}

extern "C" void kernel_launch(void* const* d_in, const int* in_sizes, int n_in,
                              void* d_out, int out_size, void* d_ws, size_t ws_size,
                              hipStream_t stream) {
    const float* x = (const float*)d_in[0];
    const int*   p = (const int*)d_in[1];
    float* out = (float*)d_out;

    dim3 grid(C_IN, B_SZ);   // one workgroup per (batch, input-channel) plane
    equiv_upsample_kernel<<<grid, 256, 0, stream>>>(x, p, out);
}